// NSAAttention_81174881895039
// MI455X (gfx1250) — compile-verified
//
#include <hip/hip_runtime.h>

typedef __attribute__((ext_vector_type(2))) float v2f;
typedef __attribute__((ext_vector_type(8))) float v8f;

#define S_LEN 2048
#define HQ 16
#define HKV 2
#define GQ 8
#define DIM 64
#define STRIDE_C 16
#define LCMP 32
#define SC 127
#define SCP 128
#define LBLK 64
#define KSEL 8
#define WIN 512
#define QK_SCALE 0.125f

__device__ inline v8f v8zero() {
  v8f x;
#pragma unroll
  for (int i = 0; i < 8; ++i) x[i] = 0.f;
  return x;
}

// D = A(16x4) * B(4x16) + C, all f32, wave32. 8-arg form per probe:
// (neg_a, A, neg_b, B, c_mod, C, reuse_a, reuse_b)
__device__ inline v8f wmma4(v2f a, v2f b, v8f c) {
  return __builtin_amdgcn_wmma_f32_16x16x4_f32(false, a, false, b, (short)0, c,
                                               false, false);
}

// reductions across the 16 lanes of a half-wave (rows r / r+8 live in halves)
__device__ inline float rowmax16(float x) {
#pragma unroll
  for (int off = 1; off < 16; off <<= 1) x = fmaxf(x, __shfl_xor(x, off, 16));
  return x;
}
__device__ inline float rowsum16(float x) {
#pragma unroll
  for (int off = 1; off < 16; off <<= 1) x += __shfl_xor(x, off, 16);
  return x;
}

// ---------------------------------------------------------------------------
// Kernel A: conv1d compression of K/V  -> ck/cv [Hkv][Sc][D]
// ck[h,sc,o] = sum_{i,t} K[sc*16+t, h, i] * wk[o,i,t]
// ---------------------------------------------------------------------------
__global__ __launch_bounds__(64) void nsa_compress(
    const float* __restrict__ k, const float* __restrict__ v,
    const float* __restrict__ wk, const float* __restrict__ wv,
    float* __restrict__ ck, float* __restrict__ cv) {
  __shared__ __align__(16) float Ks[LCMP][DIM];
  __shared__ __align__(16) float Vs[LCMP][DIM];
  const int h = blockIdx.x / SC;
  const int sc = blockIdx.x % SC;
  const int tid = threadIdx.x;
  const int s0 = sc * STRIDE_C;
  for (int e = tid; e < LCMP * DIM; e += 64) {
    int r = e >> 6, c = e & 63;
    Ks[r][c] = k[((s0 + r) * HKV + h) * DIM + c];
    Vs[r][c] = v[((s0 + r) * HKV + h) * DIM + c];
  }
  __syncthreads();
  const int d = tid;
  float ak = 0.f, av = 0.f;
  for (int din = 0; din < DIM; ++din) {
    const float* wkp = &wk[(d * DIM + din) * LCMP];
    const float* wvp = &wv[(d * DIM + din) * LCMP];
#pragma unroll 8
    for (int t = 0; t < LCMP; ++t) {
      ak = fmaf(Ks[t][din], wkp[t], ak);
      av = fmaf(Vs[t][din], wvp[t], av);
    }
  }
  ck[(h * SC + sc) * DIM + d] = ak;
  cv[(h * SC + sc) * DIM + d] = av;
}

// ---------------------------------------------------------------------------
// Kernel B: compressed-KV attention (two-pass softmax, WMMA f32 16x16x4)
// block = (hkv, 16-query tile); 8 waves = 8 GQA heads. Also emits p_kv
// (probabilities summed over the group) for the selection branch.
// ---------------------------------------------------------------------------
__global__ __launch_bounds__(256) void nsa_cmp_attn(
    const float* __restrict__ q, const float* __restrict__ ck,
    const float* __restrict__ cv, float* __restrict__ cmp_o,
    float* __restrict__ p_kv) {
  __shared__ __align__(16) float tile[16 * DIM];   // staged ck / cv tile
  __shared__ __align__(16) float pscr[8 * 256];    // per-wave 16x16 P
  __shared__ __align__(16) float pkv_s[16 * SCP];
  const int h = blockIdx.x >> 7;
  const int q0 = (blockIdx.x & 127) * 16;
  const int tid = threadIdx.x;
  const int w = tid >> 5, lane = tid & 31;
  const int n = lane & 15, hh = lane >> 4;
  const int hq = h * GQ + w;
  for (int e = tid; e < 16 * SCP; e += 256) pkv_s[e] = 0.f;

  // Q rows in A layout, pre-scaled
  v2f qa[16];
  {
    const float* qp = q + ((q0 + n) * HQ + hq) * DIM + 2 * hh;
#pragma unroll
    for (int i = 0; i < 16; ++i) {
      v2f t = *(const v2f*)(qp + 4 * i);
      qa[i] = t * QK_SCALE;
    }
  }
  v8f sacc[8];
  // pass A: all 8 key tiles of scores kept in registers (127 keys, padded)
  for (int kt = 0; kt < 8; ++kt) {
    __syncthreads();
    for (int e = tid; e < 16 * DIM; e += 256) {
      int r = e >> 6, c = e & 63;
      int j = kt * 16 + r;
      tile[e] = (j < SC) ? ck[(h * SC + j) * DIM + c] : 0.f;
    }
    __syncthreads();
    v8f c = v8zero();
    const float* bp = &tile[n * DIM + 2 * hh];
#pragma unroll
    for (int i = 0; i < 16; ++i) {
      v2f b = *(const v2f*)(bp + 4 * i);
      c = wmma4(qa[i], b, c);
    }
    sacc[kt] = c;
  }
  // full softmax stats (mask: j*16+32 <= t+1)
  float rsum[8];
#pragma unroll
  for (int r = 0; r < 8; ++r) {
    const int t = q0 + r + 8 * hh;
    float lm = -1e30f;
#pragma unroll
    for (int kt = 0; kt < 8; ++kt) {
      int j = kt * 16 + n;
      bool valid = (j < SC) && (j * STRIDE_C + LCMP <= t + 1);
      lm = fmaxf(lm, valid ? sacc[kt][r] : -1e30f);
    }
    lm = rowmax16(lm);
    float mx = (lm > -1e29f) ? lm : 0.f;
    float ps = 0.f;
#pragma unroll
    for (int kt = 0; kt < 8; ++kt) {
      int j = kt * 16 + n;
      bool valid = (j < SC) && (j * STRIDE_C + LCMP <= t + 1);
      float p = valid ? __expf(sacc[kt][r] - mx) : 0.f;
      sacc[kt][r] = p;
      ps += p;
    }
    rsum[r] = fmaxf(rowsum16(ps), 1e-20f);
  }
  // pass B: PV with normalized probabilities + p_kv accumulation
  float* pw = &pscr[w * 256];
  v8f oacc[4];
#pragma unroll
  for (int tn = 0; tn < 4; ++tn) oacc[tn] = v8zero();
  for (int kt = 0; kt < 8; ++kt) {
    __syncthreads();
    for (int e = tid; e < 16 * DIM; e += 256) {
      int r = e >> 6, c = e & 63;
      int j = kt * 16 + r;
      tile[e] = (j < SC) ? cv[(h * SC + j) * DIM + c] : 0.f;
    }
#pragma unroll
    for (int r = 0; r < 8; ++r) {
      int m = r + 8 * hh;
      float pn = sacc[kt][r] / rsum[r];
      pw[m * 16 + n] = pn;
      atomicAdd(&pkv_s[m * SCP + kt * 16 + n], pn);
    }
    __syncthreads();
#pragma unroll
    for (int tn = 0; tn < 4; ++tn) {
      const float* ap = &pw[n * 16 + 2 * hh];
#pragma unroll
      for (int i = 0; i < 4; ++i) {
        v2f a = *(const v2f*)(ap + 4 * i);
        v2f b;
        b.x = tile[(4 * i + 2 * hh) * DIM + tn * 16 + n];
        b.y = tile[(4 * i + 2 * hh + 1) * DIM + tn * 16 + n];
        oacc[tn] = wmma4(a, b, oacc[tn]);
      }
    }
  }
#pragma unroll
  for (int tn = 0; tn < 4; ++tn)
#pragma unroll
    for (int r = 0; r < 8; ++r) {
      int m = r + 8 * hh;
      cmp_o[((q0 + m) * HQ + hq) * DIM + tn * 16 + n] = oacc[tn][r];
    }
  __syncthreads();
  for (int e = tid; e < 16 * SCP; e += 256) {
    int r = e >> 7, j = e & 127;
    p_kv[(h * S_LEN + q0 + r) * SCP + j] = pkv_s[e];
  }
}

// ---------------------------------------------------------------------------
// Kernel C: pooling + forced current block + iterative top-8 (one wave/row)
// ---------------------------------------------------------------------------
__global__ __launch_bounds__(256) void nsa_select(
    const float* __restrict__ p_kv, int* __restrict__ idx_out) {
  const int w = threadIdx.x >> 5, ln = threadIdx.x & 31;
  const int row = blockIdx.x * 8 + w;  // over Hkv*S
  const int h = row >> 11, t = row & (S_LEN - 1);
  const float* pr = &p_kv[(h * S_LEN + t) * SCP];
  float val;
  if (ln < 31) {  // n_out = 31 pooled windows; block 31 padded with -1
    float s = 0.f;
#pragma unroll
    for (int tt = 0; tt < 5; ++tt) s += pr[ln * 4 + tt];
    val = s * 0.2f;
  } else {
    val = -1.0f;
  }
  if (ln == (t >> 6)) val = 3.0e38f;  // force current block
  int* op = &idx_out[(h * S_LEN + t) * KSEL];
  for (int it = 0; it < KSEL; ++it) {
    float bv = val;
    int bi = ln;
#pragma unroll
    for (int off = 16; off > 0; off >>= 1) {
      float ov = __shfl_xor(bv, off, 32);
      int oi = __shfl_xor(bi, off, 32);
      if (ov > bv || (ov == bv && oi < bi)) { bv = ov; bi = oi; }
    }
    if (ln == 0) op[it] = bi;
    if (ln == bi) val = -3.0e38f;
  }
}

// ---------------------------------------------------------------------------
// Kernel D: selected-block attention. Wave per (h, t); M=16 = 8 heads + 8
// zero pad rows (selection differs per t, so keys can't be shared). Gathered
// K/V straight from global, online softmax, WMMA f32 16x16x4.
// ---------------------------------------------------------------------------
__global__ __launch_bounds__(256) void nsa_sel_attn(
    const float* __restrict__ q, const float* __restrict__ k,
    const float* __restrict__ v, const int* __restrict__ idx_in,
    float* __restrict__ sel_o) {
  __shared__ __align__(16) float pscr[8 * 256];
  const int h = blockIdx.x >> 8;              // S/8 = 256 blocks per h
  const int tb = (blockIdx.x & 255) * 8;
  const int tid = threadIdx.x;
  const int w = tid >> 5, lane = tid & 31;
  const int n = lane & 15, hh = lane >> 4;
  const int t = tb + w;
  int sidx[8];
  {
    const int* ip = &idx_in[(h * S_LEN + t) * KSEL];
#pragma unroll
    for (int kb = 0; kb < 8; ++kb) sidx[kb] = ip[kb];
  }
  v2f qa[16];
  if (n < 8) {
    const float* qp = q + (t * HQ + h * GQ + n) * DIM + 2 * hh;
#pragma unroll
    for (int i = 0; i < 16; ++i) {
      v2f x = *(const v2f*)(qp + 4 * i);
      qa[i] = x * QK_SCALE;
    }
  } else {
#pragma unroll
    for (int i = 0; i < 16; ++i) { v2f z; z.x = 0.f; z.y = 0.f; qa[i] = z; }
  }
  float mrow[8], lrow[8];
#pragma unroll
  for (int r = 0; r < 8; ++r) { mrow[r] = -1e30f; lrow[r] = 0.f; }
  v8f oacc[4];
#pragma unroll
  for (int tn = 0; tn < 4; ++tn) oacc[tn] = v8zero();
  float* pw = &pscr[w * 256];

  for (int jt = 0; jt < 32; ++jt) {  // 512 selected keys = 32 tiles of 16
    const int base = sidx[jt >> 2] * LBLK + (jt & 3) * 16;
    const int pos_n = base + n;
    v8f c = v8zero();
    const float* kp = k + (pos_n * HKV + h) * DIM + 2 * hh;
#pragma unroll
    for (int i = 0; i < 16; ++i) {
      v2f b = *(const v2f*)(kp + 4 * i);
      c = wmma4(qa[i], b, c);
    }
    const bool valid = (pos_n <= t);
#pragma unroll
    for (int r = 0; r < 8; ++r) {
      float s = valid ? c[r] : -1e30f;
      float tm = rowmax16(s);
      float mnew = fmaxf(mrow[r], tm);
      float msafe = (mnew > -1e29f) ? mnew : 0.f;
      float cf = __expf(mrow[r] - msafe);
      float p = valid ? __expf(c[r] - msafe) : 0.f;
      lrow[r] = lrow[r] * cf + rowsum16(p);
#pragma unroll
      for (int tn = 0; tn < 4; ++tn) oacc[tn][r] *= cf;
      pw[(r + 8 * hh) * 16 + n] = p;
      mrow[r] = mnew;
    }
    __syncthreads();
#pragma unroll
    for (int tn = 0; tn < 4; ++tn) {
      const float* ap = &pw[n * 16 + 2 * hh];
#pragma unroll
      for (int i = 0; i < 4; ++i) {
        v2f a = *(const v2f*)(ap + 4 * i);
        int k0 = base + 4 * i + 2 * hh;
        v2f b;
        b.x = v[(k0 * HKV + h) * DIM + tn * 16 + n];
        b.y = v[((k0 + 1) * HKV + h) * DIM + tn * 16 + n];
        oacc[tn] = wmma4(a, b, oacc[tn]);
      }
    }
    __syncthreads();
  }
  if (hh == 0) {  // only rows 0..7 are real heads
#pragma unroll
    for (int r = 0; r < 8; ++r) {
      float inv = 1.f / fmaxf(lrow[r], 1e-20f);
#pragma unroll
      for (int tn = 0; tn < 4; ++tn)
        sel_o[(t * HQ + h * GQ + r) * DIM + tn * 16 + n] = oacc[tn][r] * inv;
    }
  }
}

// ---------------------------------------------------------------------------
// Kernel E: sliding-window attention (W=512), K/V staged through LDS and
// shared by the 8 GQA-head waves, online softmax, WMMA f32 16x16x4.
// ---------------------------------------------------------------------------
__global__ __launch_bounds__(256) void nsa_swa(
    const float* __restrict__ q, const float* __restrict__ k,
    const float* __restrict__ v, float* __restrict__ swa_o) {
  __shared__ __align__(16) float ktile[16 * DIM];
  __shared__ __align__(16) float vtile[16 * DIM];
  __shared__ __align__(16) float pscr[8 * 256];
  const int h = blockIdx.x >> 7;
  const int q0 = (blockIdx.x & 127) * 16;
  const int tid = threadIdx.x;
  const int w = tid >> 5, lane = tid & 31;
  const int n = lane & 15, hh = lane >> 4;
  const int hq = h * GQ + w;
  v2f qa[16];
  {
    const float* qp = q + ((q0 + n) * HQ + hq) * DIM + 2 * hh;
#pragma unroll
    for (int i = 0; i < 16; ++i) {
      v2f x = *(const v2f*)(qp + 4 * i);
      qa[i] = x * QK_SCALE;
    }
  }
  const int lo = q0 - (WIN - 1);
  const int kt_lo = (lo > 0) ? (lo >> 4) : 0;
  const int kt_hi = (q0 + 15) >> 4;
  float mrow[8], lrow[8];
#pragma unroll
  for (int r = 0; r < 8; ++r) { mrow[r] = -1e30f; lrow[r] = 0.f; }
  v8f oacc[4];
#pragma unroll
  for (int tn = 0; tn < 4; ++tn) oacc[tn] = v8zero();
  float* pw = &pscr[w * 256];

  for (int kt = kt_lo; kt <= kt_hi; ++kt) {
    __syncthreads();
    for (int e = tid; e < 16 * DIM; e += 256) {
      int r = e >> 6, c = e & 63;
      int j = kt * 16 + r;
      ktile[e] = k[(j * HKV + h) * DIM + c];
      vtile[e] = v[(j * HKV + h) * DIM + c];
    }
    if (kt < kt_hi) {  // hint next tile into cache (global_prefetch_b8)
      __builtin_prefetch(&k[((kt + 1) * 16 * HKV + h) * DIM + (tid & 15) * 16], 0, 1);
      __builtin_prefetch(&v[((kt + 1) * 16 * HKV + h) * DIM + (tid & 15) * 16], 0, 1);
    }
    __syncthreads();
    v8f c = v8zero();
    const float* bp = &ktile[n * DIM + 2 * hh];
#pragma unroll
    for (int i = 0; i < 16; ++i) {
      v2f b = *(const v2f*)(bp + 4 * i);
      c = wmma4(qa[i], b, c);
    }
    const int j = kt * 16 + n;
#pragma unroll
    for (int r = 0; r < 8; ++r) {
      const int t = q0 + r + 8 * hh;
      bool valid = (j <= t) && (j >= t - (WIN - 1));
      float s = valid ? c[r] : -1e30f;
      float tm = rowmax16(s);
      float mnew = fmaxf(mrow[r], tm);
      float msafe = (mnew > -1e29f) ? mnew : 0.f;
      float cf = __expf(mrow[r] - msafe);
      float p = valid ? __expf(c[r] - msafe) : 0.f;
      lrow[r] = lrow[r] * cf + rowsum16(p);
#pragma unroll
      for (int tn = 0; tn < 4; ++tn) oacc[tn][r] *= cf;
      pw[(r + 8 * hh) * 16 + n] = p;
      mrow[r] = mnew;
    }
    __syncthreads();
#pragma unroll
    for (int tn = 0; tn < 4; ++tn) {
      const float* ap = &pw[n * 16 + 2 * hh];
#pragma unroll
      for (int i = 0; i < 4; ++i) {
        v2f a = *(const v2f*)(ap + 4 * i);
        v2f b;
        b.x = vtile[(4 * i + 2 * hh) * DIM + tn * 16 + n];
        b.y = vtile[(4 * i + 2 * hh + 1) * DIM + tn * 16 + n];
        oacc[tn] = wmma4(a, b, oacc[tn]);
      }
    }
  }
#pragma unroll
  for (int r = 0; r < 8; ++r) {
    float inv = 1.f / fmaxf(lrow[r], 1e-20f);
#pragma unroll
    for (int tn = 0; tn < 4; ++tn)
      swa_o[((q0 + r + 8 * hh) * HQ + hq) * DIM + tn * 16 + n] =
          oacc[tn][r] * inv;
  }
}

// ---------------------------------------------------------------------------
// Kernel F: gate = q . wg (3 dots per (t,hq)); out = g0*sel + g1*swa + g2*cmp
// ---------------------------------------------------------------------------
__global__ __launch_bounds__(64) void nsa_fuse(
    const float* __restrict__ q, const float* __restrict__ wg,
    const float* __restrict__ sel_o, const float* __restrict__ swa_o,
    const float* __restrict__ cmp_o, float* __restrict__ out) {
  __shared__ float red[3][64];
  const int th = blockIdx.x;  // t*HQ + hq
  const int d = threadIdx.x;
  const float qd = q[th * DIM + d];
  red[0][d] = qd * wg[0 * DIM + d];
  red[1][d] = qd * wg[1 * DIM + d];
  red[2][d] = qd * wg[2 * DIM + d];
  __syncthreads();
  for (int s = 32; s > 0; s >>= 1) {
    if (d < s) {
      red[0][d] += red[0][d + s];
      red[1][d] += red[1][d + s];
      red[2][d] += red[2][d + s];
    }
    __syncthreads();
  }
  const float g0 = red[0][0], g1 = red[1][0], g2 = red[2][0];
  const int o = th * DIM + d;
  out[o] = g0 * sel_o[o] + g1 * swa_o[o] + g2 * cmp_o[o];
}

// ---------------------------------------------------------------------------
extern "C" void kernel_launch(void* const* d_in, const int* in_sizes, int n_in,
                              void* d_out, int out_size, void* d_ws,
                              size_t ws_size, hipStream_t stream) {
  (void)in_sizes; (void)n_in; (void)out_size; (void)ws_size;
  const float* q  = (const float*)d_in[0];
  const float* k  = (const float*)d_in[1];
  const float* v  = (const float*)d_in[2];
  const float* wk = (const float*)d_in[3];
  const float* wv = (const float*)d_in[4];
  const float* wg = (const float*)d_in[5];
  // d_in[6] = cu_seqlens (B=1, unused)

  float* ws    = (float*)d_ws;
  float* ck    = ws;                               // [2][127][64]
  float* cv    = ck + HKV * SC * DIM;              // [2][127][64]
  float* p_kv  = cv + HKV * SC * DIM;              // [2][2048][128]
  int*   idx   = (int*)(p_kv + HKV * S_LEN * SCP); // [2][2048][8]
  float* cmp_o = (float*)(idx + HKV * S_LEN * KSEL);
  float* sel_o = cmp_o + S_LEN * HQ * DIM;
  float* swa_o = sel_o + S_LEN * HQ * DIM;
  float* out   = (float*)d_out;

  nsa_compress<<<HKV * SC, 64, 0, stream>>>(k, v, wk, wv, ck, cv);
  nsa_cmp_attn<<<HKV * (S_LEN / 16), 256, 0, stream>>>(q, ck, cv, cmp_o, p_kv);
  nsa_select<<<(HKV * S_LEN) / 8, 256, 0, stream>>>(p_kv, idx);
  nsa_sel_attn<<<HKV * (S_LEN / 8), 256, 0, stream>>>(q, k, v, idx, sel_o);
  nsa_swa<<<HKV * (S_LEN / 16), 256, 0, stream>>>(q, k, v, swa_o);
  nsa_fuse<<<S_LEN * HQ, 64, 0, stream>>>(q, wg, sel_o, swa_o, cmp_o, out);
}